// DecoderLayer_60559038873664
// MI455X (gfx1250) — compile-verified
//
#include <hip/hip_runtime.h>
#include <math.h>

// Problem constants (match reference)
#define DIMN  1024
#define NH    16
#define HD    64
#define MLP   4096
#define BATCH 4
#define SEQ   1024

typedef __attribute__((ext_vector_type(2))) float v2f;
typedef __attribute__((ext_vector_type(8))) float v8f;

// D = A(16x4 f32) x B(4x16 f32) + C(16x16 f32), wave32 WMMA
__device__ __forceinline__ v8f wmma4(v2f a, v2f b, v8f c) {
  return __builtin_amdgcn_wmma_f32_16x16x4_f32(false, a, false, b, (short)0, c,
                                               false, false);
}

// ---------------------------------------------------------------------------
// GEMM: C[32 x 64 tile] = A[rows x K] @ W[K x N]  (+bias) (+exact GELU) (+resid)
// grid = (rows/32, N/64), block = 32 (one wave per tile).
// Two 16-row A fragments share each B fragment: 8 WMMAs per 10 loads/K-step,
// A fragment loaded as one b64 per row-tile.
// ---------------------------------------------------------------------------
__global__ __launch_bounds__(32)
void gemm_wmma_f32(const float* __restrict__ A, const float* __restrict__ W,
                   const float* __restrict__ bias, const float* __restrict__ resid,
                   float* __restrict__ C, int K, int N, int do_gelu)
{
  const int lane = threadIdx.x & 31;
  const int lr   = lane & 15;   // col within tile (B/C/D), row within tile (A)
  const int hs   = lane >> 4;   // half select: kbase = 2*hs for A/B; row+8 for C/D
  const int row0 = blockIdx.x * 32;
  const int col0 = blockIdx.y * 64;

  const float* Arow0 = A + (size_t)(row0 + lr) * K + (hs << 1);
  const float* Arow1 = Arow0 + (size_t)16 * K;
  v8f acc[2][4] = {};

  #pragma unroll 2
  for (int k = 0; k < K; k += 4) {
    const v2f a0 = *(const v2f*)(Arow0 + k);   // global_load_b64
    const v2f a1 = *(const v2f*)(Arow1 + k);
    const float* Wk = W + (size_t)(k + (hs << 1)) * N + col0 + lr;
    #pragma unroll
    for (int t = 0; t < 4; ++t) {
      v2f b;
      b.x = Wk[t * 16];
      b.y = Wk[N + t * 16];
      acc[0][t] = wmma4(a0, b, acc[0][t]);
      acc[1][t] = wmma4(a1, b, acc[1][t]);
    }
  }

  #pragma unroll
  for (int rr = 0; rr < 2; ++rr) {
    #pragma unroll
    for (int t = 0; t < 4; ++t) {
      const int col = col0 + t * 16 + lr;
      const float bv = bias ? bias[col] : 0.0f;
      #pragma unroll
      for (int r = 0; r < 8; ++r) {
        const int row = row0 + rr * 16 + r + 8 * hs;
        float v = acc[rr][t][r] + bv;
        if (do_gelu) v = 0.5f * v * (1.0f + erff(v * 0.70710678118654752f));
        if (resid) v += resid[(size_t)row * N + col];
        C[(size_t)row * N + col] = v;
      }
    }
  }
}

// ---------------------------------------------------------------------------
// Attention for one (batch, head, 16-row tile). Head n owns feature columns
// c with c%16==n (torch view(b,l,HEAD_DIM,N_HEADS) quirk). Mask applied
// BEFORE 1/sqrt(HD) scaling, context uses K (not V) per source quirk.
// grid = (SEQ/16, NH, BATCH), block = 32 (one wave)
// ---------------------------------------------------------------------------
__global__ __launch_bounds__(32)
void attn_wmma_f32(const float* __restrict__ Q, const float* __restrict__ Km,
                   float* __restrict__ O, int Mk, int causal)
{
  __shared__ float S[16 * 1024];   // 64 KB score strip: 16 query rows x Mk
  const int lane = threadIdx.x & 31;
  const int lr   = lane & 15;
  const int hs   = lane >> 4;
  const int l0   = blockIdx.x * 16;
  const int head = blockIdx.y;
  const int bidx = blockIdx.z;

  const float* Qb = Q  + (size_t)bidx * SEQ * DIMN;
  const float* Kb = Km + (size_t)bidx * Mk  * DIMN;

  // Preload Q A-fragments for the whole HD=64 (16 k-steps of 4)
  v2f qa[16];
  {
    const float* qrow = Qb + (size_t)(l0 + lr) * DIMN + head;
    #pragma unroll
    for (int c = 0; c < 16; ++c) {
      const int d0 = c * 4 + (hs << 1);
      qa[c].x = qrow[d0 * 16];
      qa[c].y = qrow[(d0 + 1) * 16];
    }
  }

  const float scale = 0.125f;  // 1/sqrt(64)
  for (int m0 = 0; m0 < Mk; m0 += 16) {
    v8f s = {};
    const float* krow = Kb + (size_t)(m0 + lr) * DIMN + head;
    #pragma unroll
    for (int c = 0; c < 16; ++c) {
      const int d0 = c * 4 + (hs << 1);
      v2f b;
      b.x = krow[d0 * 16];
      b.y = krow[(d0 + 1) * 16];
      s = wmma4(qa[c], b, s);
    }
    #pragma unroll
    for (int r = 0; r < 8; ++r) {
      const int row = r + 8 * hs;
      float v = s[r];
      if (causal && (m0 + lr) > (l0 + row)) v = -1e9f;  // mask BEFORE scale
      S[row * 1024 + m0 + lr] = v * scale;
    }
  }
  __syncthreads();

  // Exact softmax over m: lane lr handles row lr, half hs scans 512 columns
  {
    float* Srow = &S[lr * 1024 + hs * 512];
    float mx = -1e30f;
    for (int i = 0; i < 512; ++i) mx = fmaxf(mx, Srow[i]);
    mx = fmaxf(mx, __shfl_xor(mx, 16, 32));
    float sum = 0.0f;
    for (int i = 0; i < 512; ++i) {
      const float e = __expf(Srow[i] - mx);
      Srow[i] = e;
      sum += e;
    }
    sum += __shfl_xor(sum, 16, 32);
    const float inv = 1.0f / sum;
    for (int i = 0; i < 512; ++i) Srow[i] *= inv;
  }
  __syncthreads();

  // Context: O(16 x 64) = P(16 x Mk) @ K_head(Mk x 64)  (source uses K, not V)
  v8f o[4] = {};
  #pragma unroll 2
  for (int m = 0; m < Mk; m += 4) {
    const v2f pa = *(const v2f*)&S[lr * 1024 + m + (hs << 1)];  // ds_load_b64
    const float* kr = Kb + (size_t)(m + (hs << 1)) * DIMN + head;
    #pragma unroll
    for (int t = 0; t < 4; ++t) {
      v2f b;
      b.x = kr[(t * 16 + lr) * 16];
      b.y = kr[DIMN + (t * 16 + lr) * 16];
      o[t] = wmma4(pa, b, o[t]);
    }
  }
  #pragma unroll
  for (int t = 0; t < 4; ++t) {
    #pragma unroll
    for (int r = 0; r < 8; ++r) {
      const int row = l0 + r + 8 * hs;
      O[((size_t)bidx * SEQ + row) * DIMN + (t * 16 + lr) * 16 + head] = o[t][r];
    }
  }
}

// ---------------------------------------------------------------------------
// In-place LayerNorm over last dim (1024). grid = rows, block = 256.
// ---------------------------------------------------------------------------
__global__ __launch_bounds__(256)
void layernorm_kernel(float* __restrict__ X, const float* __restrict__ g,
                      const float* __restrict__ b)
{
  __shared__ float red[16];
  float* x = X + (size_t)blockIdx.x * DIMN;
  const int t = threadIdx.x;
  const int wid = t >> 5;

  float v[4];
  float s = 0.0f;
  #pragma unroll
  for (int i = 0; i < 4; ++i) { v[i] = x[t + 256 * i]; s += v[i]; }
  #pragma unroll
  for (int o = 16; o > 0; o >>= 1) s += __shfl_xor(s, o, 32);
  if ((t & 31) == 0) red[wid] = s;
  __syncthreads();
  float tot = 0.0f;
  #pragma unroll
  for (int i = 0; i < 8; ++i) tot += red[i];
  const float mean = tot * (1.0f / DIMN);

  float vs = 0.0f;
  #pragma unroll
  for (int i = 0; i < 4; ++i) { const float d = v[i] - mean; vs += d * d; }
  #pragma unroll
  for (int o = 16; o > 0; o >>= 1) vs += __shfl_xor(vs, o, 32);
  if ((t & 31) == 0) red[8 + wid] = vs;
  __syncthreads();
  float vtot = 0.0f;
  #pragma unroll
  for (int i = 0; i < 8; ++i) vtot += red[8 + i];
  const float rstd = rsqrtf(vtot * (1.0f / DIMN) + 1e-5f);

  #pragma unroll
  for (int i = 0; i < 4; ++i) {
    const int c = t + 256 * i;
    x[c] = (v[i] - mean) * rstd * g[c] + b[c];
  }
}

// ---------------------------------------------------------------------------
// Launch: decoder layer = SA + LN, CA + LN, FFN + LN. wv GEMMs skipped (v
// is dead in the source). FFN chunked by 1024 rows (workspace ~96 MB).
// ---------------------------------------------------------------------------
extern "C" void kernel_launch(void* const* d_in, const int* in_sizes, int n_in,
                              void* d_out, int out_size, void* d_ws, size_t ws_size,
                              hipStream_t stream)
{
  (void)in_sizes; (void)n_in; (void)out_size; (void)ws_size;
  const float* x     = (const float*)d_in[0];
  const float* enc   = (const float*)d_in[1];
  const float* sa_wq = (const float*)d_in[4];
  const float* sa_wk = (const float*)d_in[5];
  const float* sa_wo = (const float*)d_in[7];
  const float* ca_wq = (const float*)d_in[8];
  const float* ca_wk = (const float*)d_in[9];
  const float* ca_wo = (const float*)d_in[11];
  const float* ln1_g = (const float*)d_in[12];
  const float* ln1_b = (const float*)d_in[13];
  const float* ln2_g = (const float*)d_in[14];
  const float* ln2_b = (const float*)d_in[15];
  const float* ln3_g = (const float*)d_in[16];
  const float* ln3_b = (const float*)d_in[17];
  const float* ff_w1 = (const float*)d_in[18];
  const float* ff_b1 = (const float*)d_in[19];
  const float* ff_w2 = (const float*)d_in[20];
  const float* ff_b2 = (const float*)d_in[21];
  float* out = (float*)d_out;

  const int R = BATCH * SEQ;               // 4096 token rows
  const size_t SZ = (size_t)R * DIMN;      // elems per activation buffer
  float* Qb = (float*)d_ws;
  float* Kb = Qb + SZ;
  float* Ab = Kb + SZ;
  float* H1 = Ab + SZ;
  float* H2 = H1 + SZ;
  float* Fb = H2 + SZ;                     // 1024 x MLP chunk

  const dim3 blk(32);
  const dim3 gProj(R / 32, DIMN / 64);
  const dim3 gAtt(SEQ / 16, NH, BATCH);

  // ---- self-attention (+residual, +LN1) ----
  gemm_wmma_f32<<<gProj, blk, 0, stream>>>(x, sa_wq, nullptr, nullptr, Qb, DIMN, DIMN, 0);
  gemm_wmma_f32<<<gProj, blk, 0, stream>>>(x, sa_wk, nullptr, nullptr, Kb, DIMN, DIMN, 0);
  attn_wmma_f32<<<gAtt, blk, 0, stream>>>(Qb, Kb, Ab, SEQ, 1);
  gemm_wmma_f32<<<gProj, blk, 0, stream>>>(Ab, sa_wo, nullptr, x, H1, DIMN, DIMN, 0);
  layernorm_kernel<<<R, 256, 0, stream>>>(H1, ln1_g, ln1_b);

  // ---- cross-attention (+residual, +LN2) ----
  gemm_wmma_f32<<<gProj, blk, 0, stream>>>(H1, ca_wq, nullptr, nullptr, Qb, DIMN, DIMN, 0);
  gemm_wmma_f32<<<gProj, blk, 0, stream>>>(enc, ca_wk, nullptr, nullptr, Kb, DIMN, DIMN, 0);
  attn_wmma_f32<<<gAtt, blk, 0, stream>>>(Qb, Kb, Ab, SEQ, 0);
  gemm_wmma_f32<<<gProj, blk, 0, stream>>>(Ab, ca_wo, nullptr, H1, H2, DIMN, DIMN, 0);
  layernorm_kernel<<<R, 256, 0, stream>>>(H2, ln2_g, ln2_b);

  // ---- FFN (gelu exact) + residual, chunked by 1024 rows, then LN3 ----
  for (int c = 0; c < 4; ++c) {
    const float* Hc = H2 + (size_t)c * 1024 * DIMN;
    float* Oc = out + (size_t)c * 1024 * DIMN;
    const dim3 g1(1024 / 32, MLP / 64);
    gemm_wmma_f32<<<g1, blk, 0, stream>>>(Hc, ff_w1, ff_b1, nullptr, Fb, DIMN, MLP, 1);
    const dim3 g2(1024 / 32, DIMN / 64);
    gemm_wmma_f32<<<g2, blk, 0, stream>>>(Fb, ff_w2, ff_b2, Hc, Oc, MLP, DIMN, 0);
  }
  layernorm_kernel<<<R, 256, 0, stream>>>(out, ln3_g, ln3_b);
}